// NeuralTuringMachine_59176059404744
// MI455X (gfx1250) — compile-verified
//
#include <hip/hip_runtime.h>

// ---------------------------------------------------------------------------
// NTM forward for MI455X (gfx1250, wave32, WMMA).
// read_vector == memory broadcast (softmax over k sums to 1), Wr is unused.
// All GEMMs use V_WMMA_F32_16X16X4_F32 (full fp32 precision, reference dtype).
// ---------------------------------------------------------------------------

typedef __attribute__((ext_vector_type(2))) float v2f;
typedef __attribute__((ext_vector_type(4))) float v4f;
typedef __attribute__((ext_vector_type(8))) float v8f;

#define NB     8192
#define NIN    128
#define ND     64
#define NKCAT  192          // NIN + ND
#define BD     (NB * ND)    // 524288

__device__ __forceinline__ float sigm(float x) { return 1.0f / (1.0f + expf(-x)); }

// ---------------------------------------------------------------------------
// Prepack LSTM weights [256,192] (=[Wih|Whh]) into WMMA B-fragment order:
// pk[ct(16)][s(48)][lane(32)][e(2)] ; B[K=i, N=c] = Wcat[c, i]
// lane: N = lane&15, K = s*4 + (lane&16 ? 2:0) + e
// ---------------------------------------------------------------------------
__global__ void pack_lstm_w(const float* __restrict__ Wih,
                            const float* __restrict__ Whh,
                            float* __restrict__ pk) {
    int tid  = blockIdx.x * blockDim.x + threadIdx.x;   // 49152 total
    int ct   = tid / 3072;
    int rem  = tid - ct * 3072;
    int s    = rem >> 6;
    int q    = rem & 63;
    int lane = q >> 1;
    int e    = q & 1;
    int i    = s * 4 + ((lane & 16) ? 2 : 0) + e;
    int c    = ct * 16 + (lane & 15);
    pk[tid]  = (i < NIN) ? Wih[c * NIN + i] : Whh[c * ND + (i - NIN)];
}

__global__ void pack_bias(const float* __restrict__ bih,
                          const float* __restrict__ bhh,
                          float* __restrict__ bsum) {
    int c = threadIdx.x;                                 // 256
    bsum[c] = bih[c] + bhh[c];
}

// ---------------------------------------------------------------------------
// Prepack head weights W[k(64)][j(64)][d(64)] into B-fragment order:
// pk[j(64)][s(16)][kt(4)][lane(32)][e(2)] ; B[K=d, N=k] = W[k,j,d]
// ---------------------------------------------------------------------------
__global__ void pack_head_w(const float* __restrict__ W,
                            float* __restrict__ pk) {
    int tid  = blockIdx.x * blockDim.x + threadIdx.x;   // 262144 total
    int j    = tid >> 12;
    int rem  = tid & 4095;
    int s    = rem >> 8;
    int rem2 = rem & 255;
    int kt   = rem2 >> 6;
    int q    = rem2 & 63;
    int lane = q >> 1;
    int e    = q & 1;
    int d    = s * 4 + ((lane & 16) ? 2 : 0) + e;
    int k    = kt * 16 + (lane & 15);
    pk[tid]  = W[(k * ND + j) * ND + d];
}

// read_vector[b,:] = memory[:]  (exact: softmax over k sums to 1)
__global__ void bcast_mem(const float* __restrict__ mem, float* __restrict__ out) {
    int tid = blockIdx.x * blockDim.x + threadIdx.x;    // 524288 total
    out[tid] = mem[tid & (ND - 1)];
}

// ---------------------------------------------------------------------------
// LSTM cell. Block = 128 threads = 4 waves, one 16-row b-tile.
// Wave w owns d-tile w; its i/f/g/o gate column-tiles are ct = w, w+4, w+8, w+12.
// ---------------------------------------------------------------------------
__global__ void lstm_kernel(const float* __restrict__ x,
                            const float* __restrict__ h0,
                            const float* __restrict__ c0,
                            const float* __restrict__ pk,
                            const float* __restrict__ bsum,
                            float* __restrict__ out_h,   // hvar = c_new
                            float* __restrict__ cvar) {  // cvar = h_new
    __shared__ float aT[16 * 194];                       // [row][i], stride 194
    const int tid  = threadIdx.x;
    const int lane = tid & 31;
    const int wv   = tid >> 5;                           // d-tile 0..3
    const int b0   = (int)blockIdx.x * 16;

    for (int idx = tid; idx < 16 * NKCAT; idx += 128) {
        int r = idx / NKCAT, i = idx - r * NKCAT;
        aT[r * 194 + i] = (i < NIN) ? x[(b0 + r) * NIN + i]
                                    : h0[(b0 + r) * ND + (i - NIN)];
    }
    __syncthreads();

    const int lrow  = lane & 15;
    const int khalf = (lane & 16) ? 2 : 0;

    v8f z = {0, 0, 0, 0, 0, 0, 0, 0};
    v8f ai = z, af = z, ag = z, ao = z;

    for (int s = 0; s < 48; ++s) {
        v2f a = *(const v2f*)&aT[lrow * 194 + s * 4 + khalf];
        int base = (s * 32 + lane) * 2;
        v2f bi = *(const v2f*)&pk[(wv     ) * 3072 + base];
        v2f bf = *(const v2f*)&pk[(wv +  4) * 3072 + base];
        v2f bg = *(const v2f*)&pk[(wv +  8) * 3072 + base];
        v2f bo = *(const v2f*)&pk[(wv + 12) * 3072 + base];
        ai = __builtin_amdgcn_wmma_f32_16x16x4_f32(false, a, false, bi, (short)0, ai, false, false);
        af = __builtin_amdgcn_wmma_f32_16x16x4_f32(false, a, false, bf, (short)0, af, false, false);
        ag = __builtin_amdgcn_wmma_f32_16x16x4_f32(false, a, false, bg, (short)0, ag, false, false);
        ao = __builtin_amdgcn_wmma_f32_16x16x4_f32(false, a, false, bo, (short)0, ao, false, false);
    }

    const int d  = wv * 16 + lrow;
    const float bi = bsum[(wv     ) * 16 + lrow];
    const float bf = bsum[(wv +  4) * 16 + lrow];
    const float bg = bsum[(wv +  8) * 16 + lrow];
    const float bo = bsum[(wv + 12) * 16 + lrow];
    const int rowoff = (lane & 16) ? 8 : 0;

#pragma unroll
    for (int r = 0; r < 8; ++r) {
        int b = b0 + r + rowoff;
        float iv = sigm(ai[r] + bi);
        float fv = sigm(af[r] + bf);
        float gv = tanhf(ag[r] + bg);
        float ov = sigm(ao[r] + bo);
        float cn = fv * c0[b * ND + d] + iv * gv;
        float hn = ov * tanhf(cn);
        out_h[b * ND + d] = cn;      // hvar (output #2)
        cvar[b * ND + d]  = hn;      // cvar (workspace)
    }
}

// ---------------------------------------------------------------------------
// Write path. Block = 512 threads = 16 waves; block covers 16 b-rows x 16 j
// x all 64 k. Wave wv owns j = j0+wv; k sits on the WMMA N axis (4 tiles).
// Softmax over k = max/sum reduction across 4 tiles + 16-lane-half shuffles.
// Output transposed through LDS so the 134MB store is 64B-contiguous.
// ---------------------------------------------------------------------------
__global__ void ntm_write_kernel(const float* __restrict__ cvar,
                                 const float* __restrict__ mem,
                                 const float* __restrict__ pkw,
                                 const float* __restrict__ pks,
                                 float* __restrict__ out_mem) {
    __shared__ float cT[16 * 66];            // cvar tile [row][d], stride 66
    __shared__ float oT[16 * 16 * 17];       // [b][k][j], j-stride 17
    const int tid  = threadIdx.x;            // 0..511
    const int lane = tid & 31;
    const int wv   = tid >> 5;               // 0..15 -> j offset
    const int b0   = (int)blockIdx.x * 16;
    const int j0   = (int)blockIdx.y * 16;
    const int j    = j0 + wv;

    for (int idx = tid; idx < 16 * ND; idx += 512) {
        int r = idx >> 6, d = idx & 63;
        cT[r * 66 + d] = cvar[(b0 + r) * ND + d];
    }
    __syncthreads();

    const int lrow  = lane & 15;
    const int khalf = (lane & 16) ? 2 : 0;

    v8f z = {0, 0, 0, 0, 0, 0, 0, 0};
    v8f wsa[4], esa[4];
#pragma unroll
    for (int t = 0; t < 4; ++t) { wsa[t] = z; esa[t] = z; }

    const float* pw = pkw + j * 4096;
    const float* ps = pks + j * 4096;

    for (int s = 0; s < 16; ++s) {
        v2f a = *(const v2f*)&cT[lrow * 66 + s * 4 + khalf];
        int base = s * 256 + lane * 2;
#pragma unroll
        for (int t = 0; t < 4; ++t) {
            v2f bw = *(const v2f*)&pw[base + t * 64];
            v2f bs = *(const v2f*)&ps[base + t * 64];
            wsa[t] = __builtin_amdgcn_wmma_f32_16x16x4_f32(false, a, false, bw, (short)0, wsa[t], false, false);
            esa[t] = __builtin_amdgcn_wmma_f32_16x16x4_f32(false, a, false, bs, (short)0, esa[t], false, false);
        }
    }

    const float mj = mem[j];
    const int rowoff = (lane & 16) ? 8 : 0;

#pragma unroll
    for (int r = 0; r < 8; ++r) {
        // max over 64 k (4 tiles x 16 lanes within each wave half)
        float m = fmaxf(fmaxf(wsa[0][r], wsa[1][r]), fmaxf(wsa[2][r], wsa[3][r]));
        m = fmaxf(m, __shfl_xor(m, 1, 32));
        m = fmaxf(m, __shfl_xor(m, 2, 32));
        m = fmaxf(m, __shfl_xor(m, 4, 32));
        m = fmaxf(m, __shfl_xor(m, 8, 32));
        float e0 = expf(wsa[0][r] - m);
        float e1 = expf(wsa[1][r] - m);
        float e2 = expf(wsa[2][r] - m);
        float e3 = expf(wsa[3][r] - m);
        float ssum = e0 + e1 + e2 + e3;
        ssum += __shfl_xor(ssum, 1, 32);
        ssum += __shfl_xor(ssum, 2, 32);
        ssum += __shfl_xor(ssum, 4, 32);
        ssum += __shfl_xor(ssum, 8, 32);
        float inv = 1.0f / ssum;
        float tc  = tanhf(cT[(r + rowoff) * 66 + j]);   // tanh(cvar[b,j])
        // final = mem[j]*(1 - tanh(es)) + softmax * tanh(cvar)
        wsa[0][r] = mj * (1.0f - tanhf(esa[0][r])) + e0 * inv * tc;
        wsa[1][r] = mj * (1.0f - tanhf(esa[1][r])) + e1 * inv * tc;
        wsa[2][r] = mj * (1.0f - tanhf(esa[2][r])) + e2 * inv * tc;
        wsa[3][r] = mj * (1.0f - tanhf(esa[3][r])) + e3 * inv * tc;
    }

    for (int t = 0; t < 4; ++t) {
        __syncthreads();
#pragma unroll
        for (int r = 0; r < 8; ++r) {
            int bl = r + rowoff;
            oT[(bl * 16 + lrow) * 17 + wv] = wsa[t][r];
        }
        __syncthreads();
        // cooperative 64B-contiguous store of [16k x 16b x 16j]
        int p  = tid >> 1;
        int jh = (tid & 1) * 8;
        int bl = p >> 4;
        int kl = p & 15;
        int kg = t * 16 + kl;
        const float* src = &oT[(bl * 16 + kl) * 17 + jh];
        v4f v0, v1;
#pragma unroll
        for (int jj = 0; jj < 4; ++jj) { v0[jj] = src[jj]; v1[jj] = src[4 + jj]; }
        size_t off = (size_t)kg * BD + (size_t)(b0 + bl) * ND + (size_t)(j0 + jh);
        *(v4f*)&out_mem[off]     = v0;
        *(v4f*)&out_mem[off + 4] = v1;
    }
}

// ---------------------------------------------------------------------------
extern "C" void kernel_launch(void* const* d_in, const int* in_sizes, int n_in,
                              void* d_out, int out_size, void* d_ws, size_t ws_size,
                              hipStream_t stream) {
    const float* x   = (const float*)d_in[0];
    const float* h0  = (const float*)d_in[1];
    const float* c0  = (const float*)d_in[2];
    const float* mem = (const float*)d_in[3];
    const float* Wih = (const float*)d_in[4];
    const float* Whh = (const float*)d_in[5];
    const float* bih = (const float*)d_in[6];
    const float* bhh = (const float*)d_in[7];
    // d_in[8] = Wr : unused (read_vector == memory broadcast, exact)
    const float* Ww  = (const float*)d_in[9];
    const float* Ws  = (const float*)d_in[10];

    float* out      = (float*)d_out;
    float* out_read = out;              // [B,64]
    float* out_h    = out + BD;         // [B,64]
    float* out_mem  = out + 2 * BD;     // [64,B,64]

    float* ws      = (float*)d_ws;
    float* ws_cvar = ws;                // 524288 floats
    float* ws_bsum = ws + 524288;       // 256
    float* ws_lpk  = ws + 524544;       // 49152
    float* ws_wpk  = ws + 573696;       // 262144
    float* ws_spk  = ws + 835840;       // 262144  (total ~4.4 MB)

    pack_lstm_w<<<192, 256, 0, stream>>>(Wih, Whh, ws_lpk);
    pack_bias<<<1, 256, 0, stream>>>(bih, bhh, ws_bsum);
    pack_head_w<<<1024, 256, 0, stream>>>(Ww, ws_wpk);
    pack_head_w<<<1024, 256, 0, stream>>>(Ws, ws_spk);
    bcast_mem<<<2048, 256, 0, stream>>>(mem, out_read);
    lstm_kernel<<<512, 128, 0, stream>>>(x, h0, c0, ws_lpk, ws_bsum, out_h, ws_cvar);
    ntm_write_kernel<<<dim3(512, 4), 512, 0, stream>>>(ws_cvar, mem, ws_wpk, ws_spk, out_mem);
}